// GraphPool_25958782337117
// MI455X (gfx1250) — compile-verified
//
#include <hip/hip_runtime.h>

// GraphPool (DeepChem-style) for MI455X / gfx1250.
// Output row i = max(atoms[i], max_j atoms[adj_deg[i][j]]) for each degree
// bucket; degree-0 rows are a straight copy. Memory-bound: ~1.3 GB of traffic
// -> ~57 us floor at 23.3 TB/s. Strategy: wave32 = one wave per 128-float row,
// each lane moves one float4 (global_load_b128 / global_store_b128), output
// stores are non-temporal (TH_STORE_NT) to protect the 192 MB L2 for the
// random neighbor gathers over the 256 MB feature table.

typedef __attribute__((ext_vector_type(4))) float v4f;

__device__ __forceinline__ v4f vmax4(v4f a, v4f b) {
    v4f r;
    r.x = fmaxf(a.x, b.x);
    r.y = fmaxf(a.y, b.y);
    r.z = fmaxf(a.z, b.z);
    r.w = fmaxf(a.w, b.w);
    return r;
}

// Degree-0 bucket: pure copy. One wave (32 lanes) per row, float4 per lane.
__global__ __launch_bounds__(256) void pool_deg0(const float* __restrict__ atoms,
                                                 float* __restrict__ out, int n) {
    const int row  = blockIdx.x * (blockDim.x >> 5) + (threadIdx.x >> 5);
    const int lane = threadIdx.x & 31;
    if (row >= n) return;
    const v4f* __restrict__ av = (const v4f*)atoms;
    v4f* __restrict__ ov = (v4f*)out;
    const size_t idx = (size_t)row * 32 + lane;   // 32 float4 per 128-float row
    v4f v = av[idx];                               // RT load (row is a gather target too)
    __builtin_nontemporal_store(v, &ov[idx]);      // NT store: output never re-read
}

// Degree-DEG bucket. One wave per row. Neighbor indices are wave-uniform
// (one 512 B line broadcast); all DEG gathers issued before the reduction
// for maximum memory-level parallelism.
template <int DEG>
__global__ __launch_bounds__(256) void pool_deg(const float* __restrict__ atoms,
                                                const int* __restrict__ adj,
                                                float* __restrict__ out,
                                                int begin, int n) {
    const int row  = blockIdx.x * (blockDim.x >> 5) + (threadIdx.x >> 5);
    const int lane = threadIdx.x & 31;
    if (row >= n) return;

    const v4f* __restrict__ av = (const v4f*)atoms;
    v4f* __restrict__ ov = (v4f*)out;

    const size_t selfRow = (size_t)(begin + row);
    const size_t selfIdx = selfRow * 32 + lane;

    // Self features (RT: the same table is the gather target).
    v4f acc = av[selfIdx];

    // Wave-uniform neighbor indices.
    int nbr[DEG];
#pragma unroll
    for (int j = 0; j < DEG; ++j)
        nbr[j] = adj[(size_t)row * DEG + j];

    // Issue all gathers first (DEG independent b128 loads in flight).
    v4f vals[DEG];
#pragma unroll
    for (int j = 0; j < DEG; ++j)
        vals[j] = av[(size_t)nbr[j] * 32 + lane];

#pragma unroll
    for (int j = 0; j < DEG; ++j)
        acc = vmax4(acc, vals[j]);

    __builtin_nontemporal_store(acc, &ov[selfIdx]);
}

extern "C" void kernel_launch(void* const* d_in, const int* in_sizes, int n_in,
                              void* d_out, int out_size, void* d_ws, size_t ws_size,
                              hipStream_t stream) {
    (void)n_in; (void)d_ws; (void)ws_size; (void)out_size;

    const float* atoms = (const float*)d_in[0];
    // d_in[1] = deg_slice: not needed, bucket layout is recomputed from in_sizes.
    float* out = (float*)d_out;

    constexpr int D_FEAT = 128;
    const int total = in_sizes[0] / D_FEAT;

    int n[11], begins[11];
    int sum = 0;
    for (int deg = 1; deg <= 10; ++deg) {
        n[deg] = in_sizes[1 + deg] / deg;  // adj_deg{deg} has n*deg ints
        sum += n[deg];
    }
    n[0] = total - sum;
    begins[0] = 0;
    for (int deg = 1; deg <= 10; ++deg)
        begins[deg] = begins[deg - 1] + n[deg - 1];

    const dim3 block(256);                 // 8 wave32 per block = 8 rows
    const int rpb = 256 / 32;

    if (n[0] > 0) {
        pool_deg0<<<(n[0] + rpb - 1) / rpb, block, 0, stream>>>(atoms, out, n[0]);
    }

#define LAUNCH_DEG(DEG)                                                          \
    if (n[DEG] > 0) {                                                            \
        pool_deg<DEG><<<(n[DEG] + rpb - 1) / rpb, block, 0, stream>>>(           \
            atoms, (const int*)d_in[1 + DEG], out, begins[DEG], n[DEG]);         \
    }

    LAUNCH_DEG(1)
    LAUNCH_DEG(2)
    LAUNCH_DEG(3)
    LAUNCH_DEG(4)
    LAUNCH_DEG(5)
    LAUNCH_DEG(6)
    LAUNCH_DEG(7)
    LAUNCH_DEG(8)
    LAUNCH_DEG(9)
    LAUNCH_DEG(10)
#undef LAUNCH_DEG
}